// SSIM_36429912604808
// MI455X (gfx1250) — compile-verified
//
#include <hip/hip_runtime.h>
#include <hip/hip_bf16.h>

typedef float v2f __attribute__((ext_vector_type(2)));
typedef float v8f __attribute__((ext_vector_type(8)));

#define WPB   4      // waves per block (128 threads)
#define LSTR  34     // padded LDS row stride (floats): 16 distinct banks, keeps 8B align
#define HSTR  34     // padded stride for transposed H buffer
#define HALO  26
#define IMG   512
#define NPLANE 48    // B*C = 16*3
#define TILES_PER_PLANE 1024  // (512/16)^2

__device__ __forceinline__ float wgt_g(const float* gw, int d) {
    return (d >= 0 && d <= 10) ? gw[d] : 0.0f;
}
__device__ __forceinline__ float wgt_b(int d) {
    return (d >= 0 && d <= 10) ? (1.0f / 11.0f) : 0.0f;
}

__global__ void ssim_zero_kernel(float* acc) {
    if (threadIdx.x < 32) acc[threadIdx.x] = 0.0f;
}

__global__ __launch_bounds__(128) void ssim_main_kernel(
    const float* __restrict__ img1, const float* __restrict__ img2,
    const int* __restrict__ match, float* __restrict__ acc)
{
    __shared__ float gw[11];
    __shared__ __align__(16) float XT[WPB][32 * LSTR];
    __shared__ __align__(16) float YT[WPB][32 * LSTR];
    __shared__ __align__(16) float MT[WPB][32 * LSTR];
    __shared__ __align__(16) float HT[WPB][16 * HSTR];   // transposed H intermediate

    // Normalized 11-tap Gaussian (sigma = 1.5) built once per block.
    if (threadIdx.x == 0) {
        float t[11]; float s = 0.0f;
        for (int i = 0; i < 11; ++i) {
            float d = (float)(i - 5);
            t[i] = __expf(-(d * d) / 4.5f);   // 2*sigma^2 = 4.5
            s += t[i];
        }
        for (int i = 0; i < 11; ++i) gw[i] = t[i] / s;
    }
    __syncthreads();

    const int lane = threadIdx.x & 31;
    const int wid  = threadIdx.x >> 5;
    const int nm   = lane & 15;                 // 16-dim index for this lane (A:row, B:col)
    const int koff = (lane < 16) ? 0 : 2;       // K pair selected by lane half

    // Banded coefficient operands: G[j,n] = g[j-n]; identical register layout
    // works for both the A side (V pass) and B side (H pass).
    v2f CG[7], CB[7];
#pragma unroll
    for (int s = 0; s < 7; ++s) {
        int j0 = 4 * s + koff;
        CG[s].x = wgt_g(gw, j0 - nm);
        CG[s].y = wgt_g(gw, j0 + 1 - nm);
        CB[s].x = wgt_b(j0 - nm);
        CB[s].y = wgt_b(j0 + 1 - nm);
    }

    const int gwave = blockIdx.x * WPB + wid;
    const int plane = gwave >> 10;
    if (plane >= NPLANE) return;                 // wave-uniform; EXEC stays full
    const int tile = gwave & (TILES_PER_PLANE - 1);
    const int b  = plane / 3;
    const int ch = plane - 3 * b;
    const int r0 = (tile >> 5) << 4;
    const int c0 = (tile & 31) << 4;

    const float* p1 = img1 + (size_t)plane * (IMG * IMG);
    const float* p2 = img2 + (size_t)plane * (IMG * IMG);
    const int*   pm = match + (size_t)b * (IMG * IMG);

    float* X = XT[wid]; float* Y = YT[wid]; float* M = MT[wid]; float* H = HT[wid];

    // ---- halo load: lane owns a fixed column; row bounds are wave-uniform ----
    {
        const int cc = lane;                      // LDS column 0..31
        const int gc = c0 - 5 + cc;               // global column
        const bool colok = (cc < HALO) && (gc >= 0) && (gc < IMG);
        const int base0 = (r0 - 5) * IMG + gc;    // may be negative; loads are guarded
        const float* q1 = p1 + base0;
        const float* q2 = p2 + base0;
        const int*   qm = pm + base0;
        float* Xc = X + cc; float* Yc = Y + cc; float* Mc = M + cc;
#pragma unroll
        for (int rr = 0; rr < HALO; ++rr) {
            const int gr = r0 - 5 + rr;           // uniform per iteration
            float x = 0.0f, y = 0.0f, m = 0.0f;
            if (colok && gr >= 0 && gr < IMG) {   // immediate-offset loads under EXEC
                x = q1[rr * IMG];
                y = q2[rr * IMG];
                m = (float)qm[rr * IMG];
            }
            Xc[rr * LSTR] = x; Yc[rr * LSTR] = y; Mc[rr * LSTR] = m;
        }
#pragma unroll
        for (int rr = HALO; rr < 32; ++rr) {      // zero pad rows 26..31
            Xc[rr * LSTR] = 0.0f; Yc[rr * LSTR] = 0.0f; Mc[rr * LSTR] = 0.0f;
        }
    }
    // Single-wave tile: per-wave LDS ops complete in order; no barrier needed.

    v8f vq[6];
#pragma unroll
    for (int q = 0; q < 6; ++q) {
        // ---- horizontal pass: Data(16x28) x Gband(28x16), two row blocks ----
#pragma unroll
        for (int blk = 0; blk < 2; ++blk) {
            v8f acch = {0.f, 0.f, 0.f, 0.f, 0.f, 0.f, 0.f, 0.f};
            const int row  = nm + 16 * blk;
            const int base = row * LSTR + koff;
#pragma unroll
            for (int s = 0; s < 7; ++s) {
                const int o = base + 4 * s;
                v2f a;
                if      (q == 0) { a.x = X[o];            a.y = X[o + 1]; }
                else if (q == 1) { a.x = Y[o];            a.y = Y[o + 1]; }
                else if (q == 2) { float u = X[o], v = X[o + 1]; a.x = u * u; a.y = v * v; }
                else if (q == 3) { float u = Y[o], v = Y[o + 1]; a.x = u * u; a.y = v * v; }
                else if (q == 4) { a.x = X[o] * Y[o];     a.y = X[o + 1] * Y[o + 1]; }
                else             { a.x = M[o];            a.y = M[o + 1]; }
                v2f w = (q == 5) ? CB[s] : CG[s];
                acch = __builtin_amdgcn_wmma_f32_16x16x4_f32(
                    false, a, false, w, (short)0, acch, false, false);
            }
            // Store D (16x16) transposed into H so the V-pass B operand is a
            // contiguous 8-byte LDS load: H[col*HSTR + row].
            const int mbase = 16 * blk + ((lane < 16) ? 0 : 8);
#pragma unroll
            for (int r = 0; r < 8; r += 2) {
                v2f st; st.x = acch[r]; st.y = acch[r + 1];
                *(v2f*)&H[nm * HSTR + mbase + r] = st;
            }
        }
        // ---- vertical pass: Gband(16x28) x H(28x16) ----
        v8f accv = {0.f, 0.f, 0.f, 0.f, 0.f, 0.f, 0.f, 0.f};
        const int base2 = nm * HSTR + koff;
#pragma unroll
        for (int s = 0; s < 7; ++s) {
            v2f bv = *(const v2f*)&H[base2 + 4 * s];
            v2f w = (q == 5) ? CB[s] : CG[s];
            accv = __builtin_amdgcn_wmma_f32_16x16x4_f32(
                false, w, false, bv, (short)0, accv, false, false);
        }
        vq[q] = accv;
    }

    // ---- elementwise SSIM map in D layout + wave reduction ----
    const float C1c = 1.0e-4f;   // 0.01^2
    const float C2c = 9.0e-4f;   // 0.03^2
    float lsum = 0.0f, msum = 0.0f;
#pragma unroll
    for (int r = 0; r < 8; ++r) {
        float mu1 = vq[0][r], mu2 = vq[1][r];
        float exx = vq[2][r], eyy = vq[3][r], exy = vq[4][r];
        float mav = vq[5][r];
        float mask = ((mav + 1e-7f) > 0.5f) ? 1.0f : 0.0f;
        mask += 1e-7f;
        float mu1s = mu1 * mu1, mu2s = mu2 * mu2, m12 = mu1 * mu2;
        float s1 = exx - mu1s, s2 = eyy - mu2s, s12 = exy - m12;
        float num = (2.0f * m12 + C1c) * (2.0f * s12 + C2c);
        float den = (mu1s + mu2s + C1c) * (s1 + s2 + C2c);
        float ssim = num / den;
        lsum += (1.0f - ssim) * mask;
        msum += mask;
    }
#pragma unroll
    for (int off = 16; off >= 1; off >>= 1) {
        lsum += __shfl_xor(lsum, off, 32);
        msum += __shfl_xor(msum, off, 32);
    }
    if (lane == 0) {
        atomicAdd(&acc[b], lsum);                    // loss summed over all 3 channels
        if (ch == 0) atomicAdd(&acc[16 + b], msum);  // mask counted once per batch
    }
}

__global__ void ssim_final_kernel(const float* __restrict__ acc,
                                  float* __restrict__ out)
{
    if (threadIdx.x == 0) {
        float L = 0.0f;
        for (int b = 0; b < 16; ++b) {
            float s = acc[b];
            float m = acc[16 + b];
            L += s * (262144.0f / (m + 1e-6f));   // (H*W)/(m_b + 1e-6)
        }
        out[0] = L / 3.0f;
    }
}

extern "C" void kernel_launch(void* const* d_in, const int* in_sizes, int n_in,
                              void* d_out, int out_size, void* d_ws, size_t ws_size,
                              hipStream_t stream) {
    (void)in_sizes; (void)n_in; (void)out_size; (void)ws_size;
    const float* img1  = (const float*)d_in[0];
    const float* img2  = (const float*)d_in[1];
    const int*   match = (const int*)d_in[2];
    float* out = (float*)d_out;
    float* acc = (float*)d_ws;   // [0..15] per-batch loss sums, [16..31] per-batch mask sums

    ssim_zero_kernel<<<1, 32, 0, stream>>>(acc);

    const int totalWaves = NPLANE * TILES_PER_PLANE;   // 49152 tiles, one per wave
    const int blocks = totalWaves / WPB;               // 12288 blocks of 128 threads
    ssim_main_kernel<<<blocks, 128, 0, stream>>>(img1, img2, match, acc);

    ssim_final_kernel<<<1, 32, 0, stream>>>(acc, out);
}